// PatchQwen3MoeSparseMoeBlock_3959959847403
// MI455X (gfx1250) — compile-verified
//
#include <hip/hip_runtime.h>
#include <cstdint>
#include <cstddef>

// ---------------------------------------------------------------------------
// Qwen3 sparse MoE block for MI455X (gfx1250, wave32, WMMA).
// Routed (top-4) grouped-GEMM implementation with LDS-staged tiles:
//   router -> softmax/top4 -> bucket tokens by expert
//   -> gate/up GEMM + SwiGLU (64x16 macro-tiles, LDS staged, async-to-LDS)
//   -> down GEMM -> weighted combine.
// All matmuls use v_wmma_f32_16x16x4_f32 (fp32 in/out, matches fp32 reference).
// ---------------------------------------------------------------------------

#define T_TOKENS 1024
#define HIDDEN   1024
#define INTER    768
#define NEXP     32
#define TOPK     4
#define NASSIGN  (T_TOKENS * TOPK)   // 4096 (token, expert) assignments

#define BK       32                  // K-chunk staged in LDS per iteration
#define LDSPAD   4                   // row stride BK+4=36 floats: 16B-aligned
                                     // rows AND conflict-free wave-wide b64 reads

typedef float v2f __attribute__((ext_vector_type(2)));
typedef float v8f __attribute__((ext_vector_type(8)));

__device__ __forceinline__ v8f wmma_f32_k4(v2f a, v2f b, v8f c) {
  // (neg_a, A, neg_b, B, c_mod, C, reuse_a, reuse_b)
  return __builtin_amdgcn_wmma_f32_16x16x4_f32(false, a, false, b, (short)0, c,
                                               false, false);
}

// ---- async global -> LDS staging (CDNA5 ASYNCcnt path), with fallback ------
#if defined(__has_builtin)
#if __has_builtin(__builtin_amdgcn_global_load_async_to_lds_b128)
#define HAVE_ASYNC_LDS 1
#endif
#endif

// exact parameter types per hipcc diagnostic:
//   param0: int __attribute__((vector_size(16))) AS(1) *   (global source)
//   param1: (by symmetry) same vector type in AS(3)        (LDS destination)
typedef int v4i_vs __attribute__((vector_size(16)));
typedef __attribute__((address_space(1))) v4i_vs* g128_t;
typedef __attribute__((address_space(3))) v4i_vs* l128_t;

__device__ __forceinline__ void stage16(const float* g, float* l) {
#if defined(HAVE_ASYNC_LDS)
  // generic->AS1 is numerically identity; generic LDS addr truncates to the
  // 32-bit LDS offset (ISA aperture rule: low 32 bits).
  __builtin_amdgcn_global_load_async_to_lds_b128(
      (g128_t)(uintptr_t)g, (l128_t)(uint32_t)(uintptr_t)l, 0, 0);
#else
  *(float4*)l = *(const float4*)g;
#endif
}

__device__ __forceinline__ void stage_wait() {
#if defined(HAVE_ASYNC_LDS)
#if __has_builtin(__builtin_amdgcn_s_wait_asynccnt)
  __builtin_amdgcn_s_wait_asynccnt(0);
#else
  asm volatile("s_wait_asynccnt 0" ::: "memory");
#endif
#endif
}

// ---------------------------------------------------------------------------
// 0) zero per-expert counters
// ---------------------------------------------------------------------------
__global__ __launch_bounds__(32) void init_kernel(int* __restrict__ counts) {
  counts[threadIdx.x] = 0;
}

// ---------------------------------------------------------------------------
// 1) router logits: logits[t][e] = dot(x[t], Wr[e])   (one wave per token,
//    lane == expert; written straight into the logits region of d_out)
// ---------------------------------------------------------------------------
__global__ __launch_bounds__(128) void router_kernel(
    const float* __restrict__ x, const float* __restrict__ wr,
    float* __restrict__ logits) {
  const int t = blockIdx.x * 4 + (threadIdx.x >> 5);
  const int e = threadIdx.x & 31;
  const float4* xr = (const float4*)(x + (size_t)t * HIDDEN);
  const float4* wrow = (const float4*)(wr + (size_t)e * HIDDEN);
  float acc = 0.f;
#pragma unroll 4
  for (int d = 0; d < HIDDEN / 4; ++d) {
    float4 a = xr[d];
    float4 b = wrow[d];
    acc += a.x * b.x + a.y * b.y + a.z * b.z + a.w * b.w;
  }
  logits[t * NEXP + e] = acc;
}

// ---------------------------------------------------------------------------
// 2) softmax + top-4 + renormalize. softmax denominator cancels under the
//    top-k renormalization, so final weight = exp(l - max) / sum_top4.
// ---------------------------------------------------------------------------
__global__ __launch_bounds__(128) void topk_kernel(
    const float* __restrict__ logits, float* __restrict__ sel_w,
    int* __restrict__ sel_e, int* __restrict__ counts) {
  const int t = blockIdx.x * 128 + threadIdx.x;
  if (t >= T_TOKENS) return;
  float p[NEXP];
  float mx = -3.4e38f;
#pragma unroll
  for (int e = 0; e < NEXP; ++e) {
    p[e] = logits[t * NEXP + e];
    mx = fmaxf(mx, p[e]);
  }
#pragma unroll
  for (int e = 0; e < NEXP; ++e) p[e] = expf(p[e] - mx);

  int   sel[TOPK];
  float sw[TOPK];
  float tot = 0.f;
#pragma unroll
  for (int k = 0; k < TOPK; ++k) {
    int best = 0;
    float bv = -1.f;
#pragma unroll
    for (int e = 0; e < NEXP; ++e) {
      if (p[e] > bv) { bv = p[e]; best = e; }   // ties -> lowest index
    }
    sel[k] = best;
    sw[k] = bv;
    tot += bv;
    p[best] = -2.f;
  }
  const float inv = 1.f / tot;
#pragma unroll
  for (int k = 0; k < TOPK; ++k) {
    sel_w[t * TOPK + k] = sw[k] * inv;
    sel_e[t * TOPK + k] = sel[k];
    atomicAdd(&counts[sel[k]], 1);
  }
}

// ---------------------------------------------------------------------------
// 3) exclusive scan of counts -> expert row offsets; init cursors.
// ---------------------------------------------------------------------------
__global__ __launch_bounds__(32) void scan_kernel(const int* __restrict__ counts,
                                                  int* __restrict__ offsets,
                                                  int* __restrict__ cursors) {
  if (threadIdx.x == 0) {
    int acc = 0;
    for (int e = 0; e < NEXP; ++e) {
      offsets[e] = acc;
      acc += counts[e];
    }
    offsets[NEXP] = acc;
  }
  __syncthreads();
  cursors[threadIdx.x] = offsets[threadIdx.x];
}

// ---------------------------------------------------------------------------
// 4) fill per-expert token lists; remember each assignment's row slot.
// ---------------------------------------------------------------------------
__global__ __launch_bounds__(128) void fill_kernel(
    const int* __restrict__ sel_e, int* __restrict__ cursors,
    int* __restrict__ rows, int* __restrict__ pos_of) {
  const int t = blockIdx.x * 128 + threadIdx.x;
  if (t >= T_TOKENS) return;
#pragma unroll
  for (int k = 0; k < TOPK; ++k) {
    const int e = sel_e[t * TOPK + k];
    const int pos = atomicAdd(&cursors[e], 1);
    rows[pos] = t;
    pos_of[t * TOPK + k] = pos;
  }
}

// ---------------------------------------------------------------------------
// 5) grouped gate/up GEMM + SwiGLU.
//    Block = 128 threads (4 waves) computing a 64(M) x 16(N) macro-tile for
//    one expert bucket; gate and up share the staged A tile. K stepped in
//    LDS-staged chunks of BK=32, WMMA consumes v2f fragments from LDS.
// ---------------------------------------------------------------------------
__global__ __launch_bounds__(128) void gateup_kernel(
    const float* __restrict__ x, const float* __restrict__ gate,
    const float* __restrict__ up, const int* __restrict__ offsets,
    const int* __restrict__ rows, float* __restrict__ hbuf) {
  __shared__ float As[64][BK + LDSPAD];
  __shared__ float Bg[16][BK + LDSPAD];
  __shared__ float Bu[16][BK + LDSPAD];

  const int e = blockIdx.z;
  const int r0 = offsets[e];
  const int nrows = offsets[e + 1] - r0;
  const int mbase = blockIdx.y * 64;
  if (mbase >= nrows) return;   // empty macro-tile for this expert
  const int nbase = blockIdx.x * 16;

  const int tid = threadIdx.x;
  const int wave = tid >> 5;
  const int lane = tid & 31;
  const int half = lane >> 4;     // K-pair select (A/B), +8 rows (C/D)
  const int mn = lane & 15;

  // staging geometry: thread -> (row = tid/8 [+16*i for A], col4 = (tid&7)*4)
  const int srow = tid >> 3;            // 0..15
  const int scol = (tid & 7) * 4;       // 0,4,...,28

  // gather token ids for the 4 A-rows this thread stages (clamped tail)
  int myTok[4];
#pragma unroll
  for (int i = 0; i < 4; ++i) {
    int rr = mbase + srow + 16 * i;
    if (rr >= nrows) rr = nrows - 1;
    myTok[i] = rows[r0 + rr];
  }

  const float* __restrict__ grow = gate + ((size_t)e * INTER + nbase + srow) * HIDDEN;
  const float* __restrict__ urow = up   + ((size_t)e * INTER + nbase + srow) * HIDDEN;

  v8f cg = {};
  v8f cu = {};
  for (int k0 = 0; k0 < HIDDEN; k0 += BK) {
    __syncthreads();   // previous chunk's LDS reads complete across waves
    // stage A (64 x BK), coalesced 16B per thread per row
#pragma unroll
    for (int i = 0; i < 4; ++i) {
      stage16(x + (size_t)myTok[i] * HIDDEN + k0 + scol,
              &As[srow + 16 * i][scol]);
    }
    // stage Bg / Bu (16 x BK each)
    stage16(grow + k0 + scol, &Bg[srow][scol]);
    stage16(urow + k0 + scol, &Bu[srow][scol]);
    stage_wait();      // this wave's async copies landed in LDS
    __syncthreads();   // all waves' stages visible

#pragma unroll
    for (int k = 0; k < BK; k += 4) {
      const int kk = k + half * 2;
      v2f a  = *(const v2f*)&As[wave * 16 + mn][kk];
      v2f bg = *(const v2f*)&Bg[mn][kk];
      v2f bu = *(const v2f*)&Bu[mn][kk];
      cg = wmma_f32_k4(a, bg, cg);
      cu = wmma_f32_k4(a, bu, cu);
    }
  }

  const int mtile = mbase + wave * 16;
#pragma unroll
  for (int r = 0; r < 8; ++r) {
    const int m = r + 8 * half;   // C/D layout: VGPR r, half-wave -> M = r+8*half
    if (mtile + m < nrows) {
      const float g = cg[r];
      const float u = cu[r];
      const float hv = (g / (1.f + expf(-g))) * u;   // silu(g) * u
      hbuf[(size_t)(r0 + mtile + m) * INTER + nbase + mn] = hv;
    }
  }
}

// ---------------------------------------------------------------------------
// 6) grouped down GEMM: y[row, d] = h[row, :] . down[e, d, :]   (K = 768)
//    Same 64x16 macro-tile scheme; y per assignment slot -> no atomics.
// ---------------------------------------------------------------------------
__global__ __launch_bounds__(128) void down_kernel(
    const float* __restrict__ hbuf, const float* __restrict__ down,
    const int* __restrict__ offsets, float* __restrict__ ybuf) {
  __shared__ float As[64][BK + LDSPAD];
  __shared__ float Bd[16][BK + LDSPAD];

  const int e = blockIdx.z;
  const int r0 = offsets[e];
  const int nrows = offsets[e + 1] - r0;
  const int mbase = blockIdx.y * 64;
  if (mbase >= nrows) return;
  const int dbase = blockIdx.x * 16;

  const int tid = threadIdx.x;
  const int wave = tid >> 5;
  const int lane = tid & 31;
  const int half = lane >> 4;
  const int mn = lane & 15;

  const int srow = tid >> 3;
  const int scol = (tid & 7) * 4;

  // h row pointers for the 4 A-rows this thread stages (clamped tail)
  const float* hsrc[4];
#pragma unroll
  for (int i = 0; i < 4; ++i) {
    int rr = mbase + srow + 16 * i;
    if (rr >= nrows) rr = nrows - 1;
    hsrc[i] = hbuf + (size_t)(r0 + rr) * INTER;
  }
  const float* __restrict__ drow = down + ((size_t)e * HIDDEN + dbase + srow) * INTER;

  v8f c = {};
  for (int k0 = 0; k0 < INTER; k0 += BK) {
    __syncthreads();
#pragma unroll
    for (int i = 0; i < 4; ++i) {
      stage16(hsrc[i] + k0 + scol, &As[srow + 16 * i][scol]);
    }
    stage16(drow + k0 + scol, &Bd[srow][scol]);
    stage_wait();
    __syncthreads();

#pragma unroll
    for (int k = 0; k < BK; k += 4) {
      const int kk = k + half * 2;
      v2f a = *(const v2f*)&As[wave * 16 + mn][kk];
      v2f b = *(const v2f*)&Bd[mn][kk];
      c = wmma_f32_k4(a, b, c);
    }
  }

  const int mtile = mbase + wave * 16;
#pragma unroll
  for (int r = 0; r < 8; ++r) {
    const int m = r + 8 * half;
    if (mtile + m < nrows) {
      ybuf[(size_t)(r0 + mtile + m) * HIDDEN + dbase + mn] = c[r];
    }
  }
}

// ---------------------------------------------------------------------------
// 7) weighted combine: out[t, d] = sum_k w[t,k] * y[pos(t,k), d]
//    Fully overwrites out -> no zero-init of d_out needed; deterministic.
// ---------------------------------------------------------------------------
__global__ __launch_bounds__(256) void combine_kernel(
    const float* __restrict__ ybuf, const float* __restrict__ sel_w,
    const int* __restrict__ pos_of, float* __restrict__ out) {
  const int idx = blockIdx.x * 256 + threadIdx.x;   // idx = t*HIDDEN + d
  if (idx >= T_TOKENS * HIDDEN) return;
  const int t = idx >> 10;        // HIDDEN == 1024
  const int d = idx & (HIDDEN - 1);
  float acc = 0.f;
#pragma unroll
  for (int k = 0; k < TOPK; ++k) {
    const float w = sel_w[t * TOPK + k];
    const int pos = pos_of[t * TOPK + k];
    acc += w * ybuf[(size_t)pos * HIDDEN + d];
  }
  out[idx] = acc;
}

// ---------------------------------------------------------------------------
// launcher
// ---------------------------------------------------------------------------
extern "C" void kernel_launch(void* const* d_in, const int* in_sizes, int n_in,
                              void* d_out, int out_size, void* d_ws,
                              size_t ws_size, hipStream_t stream) {
  const float* x    = (const float*)d_in[0];  // [2,512,1024]
  const float* wr   = (const float*)d_in[1];  // [32,1024]
  const float* gate = (const float*)d_in[2];  // [32,768,1024]
  const float* up   = (const float*)d_in[3];  // [32,768,1024]
  const float* down = (const float*)d_in[4];  // [32,1024,768]

  float* out    = (float*)d_out;                       // [1024,1024]
  float* logits = out + (size_t)T_TOKENS * HIDDEN;     // [1024,32]

  // workspace layout (bytes)
  char* ws = (char*)d_ws;
  int*   counts  = (int*)(ws + 0);          // 32 ints
  int*   offsets = (int*)(ws + 512);        // 33 ints
  int*   cursors = (int*)(ws + 1024);       // 32 ints
  int*   sel_e   = (int*)(ws + 4096);       // 4096 ints
  int*   pos_of  = (int*)(ws + 20480);      // 4096 ints
  float* sel_w   = (float*)(ws + 36864);    // 4096 floats
  int*   rows    = (int*)(ws + 53248);      // 4096 ints
  float* hbuf    = (float*)(ws + 69632);    // 4096*768 floats (12.6 MB)
  float* ybuf    = (float*)(ws + 69632 + (size_t)NASSIGN * INTER * 4); // 4096*1024 floats

  init_kernel<<<1, 32, 0, stream>>>(counts);
  router_kernel<<<T_TOKENS / 4, 128, 0, stream>>>(x, wr, logits);
  topk_kernel<<<T_TOKENS / 128, 128, 0, stream>>>(logits, sel_w, sel_e, counts);
  scan_kernel<<<1, 32, 0, stream>>>(counts, offsets, cursors);
  fill_kernel<<<T_TOKENS / 128, 128, 0, stream>>>(sel_e, cursors, rows, pos_of);

  {
    dim3 grid(INTER / 16, T_TOKENS / 64, NEXP);   // (48, 16, 32)
    gateup_kernel<<<grid, 128, 0, stream>>>(x, gate, up, offsets, rows, hbuf);
  }
  {
    dim3 grid(HIDDEN / 16, T_TOKENS / 64, NEXP);  // (64, 16, 32)
    down_kernel<<<grid, 128, 0, stream>>>(hbuf, down, offsets, ybuf);
  }
  combine_kernel<<<(T_TOKENS * HIDDEN) / 256, 256, 0, stream>>>(ybuf, sel_w,
                                                                pos_of, out);
}